// LocalDownSample_22436909154642
// MI455X (gfx1250) — compile-verified
//
#include <hip/hip_runtime.h>
#include <math.h>

// Problem constants (fixed by setup_inputs in the reference)
constexpr int Bb = 8;     // batch
constexpr int Cc = 128;   // channels
constexpr int Nn = 2048;  // points
constexpr int Kn = 32;    // neighbors
constexpr int Mm = 1024;  // npts_ds

typedef __attribute__((ext_vector_type(2))) float v2f;
typedef __attribute__((ext_vector_type(8))) float v8f;

// D(16x16 f32) = A(16x4 f32) * B(4x16 f32) + C  — native fp32 WMMA on CDNA5
__device__ inline v8f wmma_f32_4(v2f a, v2f b, v8f c) {
  return __builtin_amdgcn_wmma_f32_16x16x4_f32(false, a, false, b, (short)0, c,
                                               false, false);
}

__device__ inline float wred_add(float v) {
  for (int o = 16; o; o >>= 1) v += __shfl_xor(v, o, 32);
  return v;
}
__device__ inline float wred_max(float v) {
  for (int o = 16; o; o >>= 1) v = fmaxf(v, __shfl_xor(v, o, 32));
  return v;
}

// ---------------- Kernel 1: AmatT[o,c] = sum_j Wq[j,c] * Wk[j,o] ----------------
// (transposed store so k_P's B-fragments are contiguous along K)
__global__ __launch_bounds__(32) void k_A(const float* __restrict__ Wq,
                                          const float* __restrict__ Wk,
                                          float* __restrict__ AmatT) {
  const int c0 = blockIdx.x << 4, c20 = blockIdx.y << 4;
  const int lane = threadIdx.x & 31, half = lane >> 4, col = lane & 15;
  v8f acc = {};
  for (int ko = 0; ko < Cc; ko += 4) {
    const int kk = ko + 2 * half;
    v2f a, b;
    a.x = Wq[kk * Cc + c0 + col];        // A[m=col][k=kk] = Wq[kk, c0+col]
    a.y = Wq[(kk + 1) * Cc + c0 + col];
    b.x = Wk[kk * Cc + c20 + col];       // B[k=kk][n=col] = Wk[kk, c20+col]
    b.y = Wk[(kk + 1) * Cc + c20 + col];
    acc = wmma_f32_4(a, b, acc);
  }
#pragma unroll
  for (int j = 0; j < 8; ++j) {
    const int row = j + (half << 3);     // row = c index, col = o index
    AmatT[(c20 + col) * Cc + (c0 + row)] = acc[j];
  }
}

// ---------------- Kernel 2: sq[b,n] = ||x[b,:,n]||^2 ----------------
__global__ __launch_bounds__(256) void k_sq(const float* __restrict__ x,
                                            float* __restrict__ sq) {
  const int g = blockIdx.x * 256 + threadIdx.x;  // B*N = 16384
  const int b = g >> 11, n = g & 2047;
  const float* xb = x + (size_t)b * Cc * Nn;
  float acc = 0.f;
  for (int c = 0; c < Cc; ++c) {
    const float v = xb[(size_t)c * Nn + n];
    acc += v * v;
  }
  sq[g] = acc;
}

// ---------------- Kernel 3: KNN via WMMA Gram + chunked top-32 ----------------
// One workgroup (8 waves) per (b, 16-row n-tile). A-fragments (the n-tile)
// are hoisted into 64 VGPRs and reused for all 128 m-tiles; each ko step
// feeds 4 independent accumulators (4 m-tiles) -> 2 vmem per wmma and 4
// parallel WMMA chains. score = 2*G[n,m] - sq[m] (row-constant -sq[n] dropped:
// rank-invariant). Top-32 per 512-col chunk -> 128 candidates -> final top-32
// with (value desc, index asc) tie-break matching jax.lax.top_k.
__global__ __launch_bounds__(256) void k_knn(const float* __restrict__ x,
                                             const float* __restrict__ sq,
                                             int* __restrict__ knn) {
  __shared__ float sc[16][512];   // 32 KB
  __shared__ float cv[16][128];   // 8 KB
  __shared__ int ci[16][128];     // 8 KB
  const int wg = blockIdx.x;      // 0..1023
  const int b = wg >> 7;
  const int n0 = (wg & 127) << 4;
  const int wid = threadIdx.x >> 5;
  const int lane = threadIdx.x & 31;
  const int half = lane >> 4, col = lane & 15;
  const float* xb = x + (size_t)b * Cc * Nn;
  const float* sqb = sq + b * Nn;
  const float NEG = -3.0e38f;

  // Hoist A fragments for the whole K=128 reduction (32 x v2f = 64 VGPRs)
  v2f afrag[32];
#pragma unroll
  for (int i = 0; i < 32; ++i) {
    const int kk = 4 * i + 2 * half;
    afrag[i].x = xb[(size_t)kk * Nn + n0 + col];
    afrag[i].y = xb[(size_t)(kk + 1) * Nn + n0 + col];
  }

#pragma unroll 1
  for (int chunk = 0; chunk < 4; ++chunk) {
    const int mchunk = chunk << 9;
    const int mbase = mchunk + (wid << 6);  // this wave's 64 columns
    v8f acc[4] = {};
#pragma unroll
    for (int i = 0; i < 32; ++i) {
      const int kk = 4 * i + 2 * half;
      const float* r0 = xb + (size_t)kk * Nn + mbase + col;
      const float* r1 = xb + (size_t)(kk + 1) * Nn + mbase + col;
      v2f b0 = {r0[0], r1[0]};
      v2f b1 = {r0[16], r1[16]};
      v2f b2 = {r0[32], r1[32]};
      v2f b3 = {r0[48], r1[48]};
      acc[0] = wmma_f32_4(afrag[i], b0, acc[0]);
      acc[1] = wmma_f32_4(afrag[i], b1, acc[1]);
      acc[2] = wmma_f32_4(afrag[i], b2, acc[2]);
      acc[3] = wmma_f32_4(afrag[i], b3, acc[3]);
    }
#pragma unroll
    for (int t = 0; t < 4; ++t) {
      const int mloc = (wid << 6) + (t << 4);
      const float sqm = sqb[mchunk + mloc + col];
#pragma unroll
      for (int j = 0; j < 8; ++j) {
        const int row = j + (half << 3);
        sc[row][mloc + col] = 2.0f * acc[t][j] - sqm;
      }
    }
    __syncthreads();
    // select top-32 of this chunk; wave wid owns rows 2*wid, 2*wid+1
    for (int rr = 0; rr < 2; ++rr) {
      const int row = (wid << 1) + rr;
      for (int s = 0; s < 32; ++s) {
        float bv = NEG;
        int bi = 0x7fffffff;
        for (int m = lane; m < 512; m += 32) {
          const float v = sc[row][m];
          if (v > bv || (v == bv && m < bi)) { bv = v; bi = m; }
        }
        for (int off = 16; off; off >>= 1) {
          const float ov = __shfl_xor(bv, off, 32);
          const int oi = __shfl_xor(bi, off, 32);
          if (ov > bv || (ov == bv && oi < bi)) { bv = ov; bi = oi; }
        }
        if (lane == 0) {
          cv[row][(chunk << 5) + s] = bv;
          ci[row][(chunk << 5) + s] = mchunk + bi;
        }
        // only lane (bi&31) ever re-reads sc[row][bi] (stride-32 scan)
        if (lane == (bi & 31)) sc[row][bi] = NEG;
      }
    }
    __syncthreads();
  }
  // final merge over 128 candidates per row
  for (int rr = 0; rr < 2; ++rr) {
    const int row = (wid << 1) + rr;
    int* out = knn + ((size_t)(b * Nn + n0 + row)) * Kn;
    for (int s = 0; s < 32; ++s) {
      float bv = -3.3e38f;
      int bi = 0x7fffffff, bp = 0;
      for (int p = lane; p < 128; p += 32) {
        const float v = cv[row][p];
        const int iv = ci[row][p];
        if (v > bv || (v == bv && iv < bi)) { bv = v; bi = iv; bp = p; }
      }
      for (int off = 16; off; off >>= 1) {
        const float ov = __shfl_xor(bv, off, 32);
        const int oi = __shfl_xor(bi, off, 32);
        const int op = __shfl_xor(bp, off, 32);
        if (ov > bv || (ov == bv && oi < bi)) { bv = ov; bi = oi; bp = op; }
      }
      if (lane == 0) out[s] = bi;
      if (lane == (bp & 31)) cv[row][bp] = -3.3e38f;
    }
  }
}

// ---------------- Kernel 4: P = X^T * Amat  [B,N,C] ----------------
// B-fragments (AmatT rows, contiguous along K -> global_load_b64) hoisted
// into 64 VGPRs; each wave computes 2 n-tiles against its o-tile.
__global__ __launch_bounds__(256) void k_P(const float* __restrict__ x,
                                           const float* __restrict__ AmatT,
                                           float* __restrict__ P) {
  const int wg = blockIdx.x;             // 512 = B * (N/32)
  const int b = wg >> 6, n0 = (wg & 63) << 5;
  const int wid = threadIdx.x >> 5, lane = threadIdx.x & 31;
  const int half = lane >> 4, col = lane & 15;
  const int o0 = wid << 4;
  const float* xb = x + (size_t)b * Cc * Nn;
  v2f bfrag[32];
#pragma unroll
  for (int i = 0; i < 32; ++i) {
    const int kk = 4 * i + 2 * half;
    bfrag[i] = *(const v2f*)(AmatT + (o0 + col) * Cc + kk);  // b64 load
  }
  v8f acc[2] = {};
#pragma unroll
  for (int i = 0; i < 32; ++i) {
    const int kk = 4 * i + 2 * half;
    const float* r0 = xb + (size_t)kk * Nn + n0 + col;
    const float* r1 = xb + (size_t)(kk + 1) * Nn + n0 + col;
    v2f a0 = {r0[0], r1[0]};
    v2f a1 = {r0[16], r1[16]};
    acc[0] = wmma_f32_4(a0, bfrag[i], acc[0]);
    acc[1] = wmma_f32_4(a1, bfrag[i], acc[1]);
  }
#pragma unroll
  for (int t = 0; t < 2; ++t)
#pragma unroll
    for (int j = 0; j < 8; ++j) {
      const int row = j + (half << 3);
      P[((size_t)(b * Nn) + n0 + (t << 4) + row) * Cc + o0 + col] = acc[t][j];
    }
}

// ------- Kernel 5: attention (softmax over K) + selection std, wave/point -------
__global__ __launch_bounds__(256) void k_attn(const float* __restrict__ x,
                                              const float* __restrict__ P,
                                              const int* __restrict__ knn,
                                              float* __restrict__ attn,
                                              float* __restrict__ sel) {
  const int p = blockIdx.x * 8 + (threadIdx.x >> 5);  // 0..16383
  const int b = p >> 11, n = p & 2047;
  const int lane = threadIdx.x & 31;
  const float* xb = x + (size_t)b * Cc * Nn;
  const float* Pn = P + ((size_t)(b * Nn) + n) * Cc;
  float ps = 0.f;
  for (int c = lane; c < Cc; c += 32) ps += Pn[c] * xb[(size_t)c * Nn + n];
  const float e_self = wred_add(ps);
  const int idx = knn[((size_t)(b * Nn) + n) * Kn + lane];  // lane = neighbor k
  float e = 0.f;
  for (int c = 0; c < Cc; ++c) e += Pn[c] * xb[(size_t)c * Nn + idx];
  e = (e - e_self) * 0.08838834764831845f;  // 1/sqrt(128)
  const float mx = wred_max(e);
  const float ex = __expf(e - mx);
  const float s = wred_add(ex);
  const float a = ex / s;
  attn[((size_t)(b * Nn) + n) * Kn + lane] = a;
  const float ss = wred_add(a * a);
  if (lane == 0)
    sel[b * Nn + n] =
        sqrtf(fmaxf(ss * (1.0f / 32.0f) - (1.0f / 1024.0f), 0.f));  // biased std
}

// ------- Kernel 6: per-batch bitonic sort (desc, index-asc ties) -> top-1024 -------
__global__ __launch_bounds__(1024) void k_sort(const float* __restrict__ sel,
                                               int* __restrict__ topi) {
  __shared__ float v[2048];
  __shared__ int id[2048];
  const int b = blockIdx.x, t = threadIdx.x;
  for (int i = t; i < 2048; i += 1024) { v[i] = sel[b * 2048 + i]; id[i] = i; }
  __syncthreads();
  for (int k = 2; k <= 2048; k <<= 1) {
    for (int j = k >> 1; j > 0; j >>= 1) {
      for (int i = t; i < 2048; i += 1024) {
        const int ixj = i ^ j;
        if (ixj > i) {
          const bool desc = ((i & k) == 0);
          const float va = v[i], vb = v[ixj];
          const int ia = id[i], ib = id[ixj];
          const bool agtb = (va > vb) || (va == vb && ia < ib);
          if (desc ? !agtb : agtb) {
            v[i] = vb; v[ixj] = va; id[i] = ib; id[ixj] = ia;
          }
        }
      }
      __syncthreads();
    }
  }
  if (t < Mm) topi[b * Mm + t] = id[t];
}

// ------- Kernel 7: S[b,m,c] = sum_k attn*(x[:,idx_k] - x[:,n]), wave/point -------
__global__ __launch_bounds__(256) void k_S(const float* __restrict__ x,
                                           const float* __restrict__ attn,
                                           const int* __restrict__ knn,
                                           const int* __restrict__ topi,
                                           float* __restrict__ S) {
  const int p = blockIdx.x * 8 + (threadIdx.x >> 5);  // 0..8191
  const int b = p >> 10, m = p & 1023;
  const int lane = threadIdx.x & 31;
  const int n = topi[b * Mm + m];
  const float* xb = x + (size_t)b * Cc * Nn;
  const float* at = attn + ((size_t)(b * Nn) + n) * Kn;
  const int* kn = knn + ((size_t)(b * Nn) + n) * Kn;
  float acc[4] = {0.f, 0.f, 0.f, 0.f};
  for (int k = 0; k < Kn; ++k) {
    const float a = at[k];
    const int idx = kn[k];
#pragma unroll
    for (int tt = 0; tt < 4; ++tt) {
      const int c = lane + (tt << 5);
      acc[tt] += a * (xb[(size_t)c * Nn + idx] - xb[(size_t)c * Nn + n]);
    }
  }
#pragma unroll
  for (int tt = 0; tt < 4; ++tt)
    S[((size_t)(b * Mm) + m) * Cc + lane + (tt << 5)] = acc[tt];
}

// ---------------- Kernel 8: Out[b,o,m] = sum_c Wv[o,c]*S[b,m,c] (WMMA) ----------------
// Both fragments contiguous along K -> b64 loads.
__global__ __launch_bounds__(256) void k_out(const float* __restrict__ Wv,
                                             const float* __restrict__ S,
                                             float* __restrict__ out) {
  const int wg = blockIdx.x;  // 512
  const int b = wg >> 6, m0 = (wg & 63) << 4;
  const int wid = threadIdx.x >> 5, lane = threadIdx.x & 31;
  const int half = lane >> 4, col = lane & 15;
  const int o0 = wid << 4;
  const float* Sb = S + (size_t)(b * Mm) * Cc;
  v8f acc = {};
  for (int ko = 0; ko < Cc; ko += 4) {
    const int kk = ko + 2 * half;
    v2f a = *(const v2f*)(Wv + (o0 + col) * Cc + kk);            // b64 load
    v2f bfr = *(const v2f*)(Sb + (size_t)(m0 + col) * Cc + kk);  // b64 load
    acc = wmma_f32_4(a, bfr, acc);
  }
#pragma unroll
  for (int j = 0; j < 8; ++j) {
    const int row = j + (half << 3);
    out[(size_t)b * Cc * Mm + (size_t)(o0 + row) * Mm + m0 + col] = acc[j];
  }
}

extern "C" void kernel_launch(void* const* d_in, const int* in_sizes, int n_in,
                              void* d_out, int out_size, void* d_ws,
                              size_t ws_size, hipStream_t stream) {
  const float* x = (const float*)d_in[0];
  const float* Wq = (const float*)d_in[1];
  const float* Wk = (const float*)d_in[2];
  const float* Wv = (const float*)d_in[3];
  float* out = (float*)d_out;

  char* ws = (char*)d_ws;
  size_t off = 0;
  float* AmatT = (float*)(ws + off); off += (size_t)Cc * Cc * 4;
  float* sq    = (float*)(ws + off); off += (size_t)Bb * Nn * 4;
  int*   knn   = (int*)(ws + off);   off += (size_t)Bb * Nn * Kn * 4;
  float* P     = (float*)(ws + off); off += (size_t)Bb * Nn * Cc * 4;
  float* attn  = (float*)(ws + off); off += (size_t)Bb * Nn * Kn * 4;
  float* sel   = (float*)(ws + off); off += (size_t)Bb * Nn * 4;
  int*   topi  = (int*)(ws + off);   off += (size_t)Bb * Mm * 4;
  float* S     = (float*)(ws + off); off += (size_t)Bb * Mm * Cc * 4;
  (void)ws_size; (void)in_sizes; (void)n_in; (void)out_size;

  k_A<<<dim3(8, 8), 32, 0, stream>>>(Wq, Wk, AmatT);
  k_sq<<<64, 256, 0, stream>>>(x, sq);
  k_knn<<<1024, 256, 0, stream>>>(x, sq, knn);
  k_P<<<512, 256, 0, stream>>>(x, AmatT, P);
  k_attn<<<2048, 256, 0, stream>>>(x, P, knn, attn, sel);
  k_sort<<<8, 1024, 0, stream>>>(sel, topi);
  k_S<<<1024, 256, 0, stream>>>(x, attn, knn, topi, S);
  k_out<<<512, 256, 0, stream>>>(Wv, S, out);
}